// CenterLoss_25434796327381
// MI455X (gfx1250) — compile-verified
//
#include <hip/hip_runtime.h>

// CenterLoss: mean over i of clamp(||x_i - centers[labels_i]||^2, 1e-12, 1e12)
// Memory-bound (~64 MB traffic -> ~2.8us at 23.3 TB/s). Cross-term + squares
// computed as diagonal of diff*diff^T via V_WMMA_F32_16X16X4_F32 (full f32).

#define BATCH 4096
#define FEAT  2048
#define WAVES_PER_BLOCK 8
#define KCHUNK (FEAT / WAVES_PER_BLOCK)   // 256 columns per wave

typedef float v2f __attribute__((ext_vector_type(2)));
typedef float v8f __attribute__((ext_vector_type(8)));

__global__ __launch_bounds__(256) void
CenterLoss_dist_kernel(const float* __restrict__ x,
                       const int*   __restrict__ labels,
                       const float* __restrict__ centers,
                       float*       __restrict__ blockSums)
{
    __shared__ float partial[WAVES_PER_BLOCK][16];
    __shared__ float rowDist[16];

    const int lane  = threadIdx.x & 31;
    const int wave  = threadIdx.x >> 5;
    const int row16 = lane & 15;          // row within 16-row tile
    const int hi    = lane >> 4;          // 0: K pair {k,k+1}; 1: {k+2,k+3}

    const int row = blockIdx.x * 16 + row16;
    const int lbl = labels[row];

    const float* __restrict__ xr = x       + (size_t)row * FEAT;
    const float* __restrict__ cr = centers + (size_t)lbl * FEAT;

    const int kBase = wave * KCHUNK + 2 * hi;

    v8f acc = {};
    #pragma unroll 8
    for (int k = 0; k < KCHUNK; k += 4) {
        const int kk = kBase + k;
        // One WGP-scope prefetch per 128B cacheline, ~1KB ahead of the
        // per-lane stream (constant-folds under the 8x unroll).
        if ((k & 31) == 0) {
            __builtin_prefetch(xr + kk + 256, 0, 3);
            __builtin_prefetch(cr + kk + 256, 0, 3);
        }
        v2f a = *(const v2f*)(xr + kk);
        v2f c = *(const v2f*)(cr + kk);
        v2f d = a - c;
        // D = diff * diff^T + C ; A and B operand register images coincide
        // for this symmetric product under the ISA 16x4 / 4x16 f32 layouts.
        acc = __builtin_amdgcn_wmma_f32_16x16x4_f32(
                  /*neg_a=*/false, d, /*neg_b=*/false, d,
                  /*c_mod=*/(short)0, acc,
                  /*reuse_a=*/false, /*reuse_b=*/false);
    }

    // Diagonal extraction per ISA C/D layout:
    //   m in 0..7  -> lane m,      acc[m]
    //   m in 8..15 -> lane m+16,   acc[m-8]   (i.e. lanes 24..31)
    const bool owner = (lane < 8) || (lane >= 24);
    const int  sel   = (lane < 8) ? lane : (lane - 24);
    float val = acc[0];
    if (sel == 1) val = acc[1];
    if (sel == 2) val = acc[2];
    if (sel == 3) val = acc[3];
    if (sel == 4) val = acc[4];
    if (sel == 5) val = acc[5];
    if (sel == 6) val = acc[6];
    if (sel == 7) val = acc[7];
    if (owner) {
        const int r = (lane < 8) ? lane : (lane - 16);
        partial[wave][r] = val;     // unique (wave,row) slot, no conflicts
    }
    __syncthreads();

    // Fixed-order combine across waves (deterministic), clamp per row.
    if (threadIdx.x < 16) {
        float dsum = 0.0f;
        #pragma unroll
        for (int w = 0; w < WAVES_PER_BLOCK; ++w)
            dsum += partial[w][threadIdx.x];
        dsum = fminf(fmaxf(dsum, 1e-12f), 1e12f);
        rowDist[threadIdx.x] = dsum;
    }
    __syncthreads();

    if (threadIdx.x == 0) {
        float s = 0.0f;
        #pragma unroll
        for (int i = 0; i < 16; ++i) s += rowDist[i];
        blockSums[blockIdx.x] = s;
    }
}

__global__ __launch_bounds__(256) void
CenterLoss_reduce_kernel(const float* __restrict__ blockSums,
                         float*       __restrict__ out)
{
    __shared__ float s[256];
    const int t = threadIdx.x;
    s[t] = blockSums[t];
    __syncthreads();
    #pragma unroll
    for (int stride = 128; stride > 0; stride >>= 1) {
        if (t < stride) s[t] += s[t + stride];
        __syncthreads();
    }
    if (t == 0) out[0] = s[0] * (1.0f / (float)BATCH);
}

extern "C" void kernel_launch(void* const* d_in, const int* in_sizes, int n_in,
                              void* d_out, int out_size, void* d_ws, size_t ws_size,
                              hipStream_t stream) {
    const float* x       = (const float*)d_in[0];
    const int*   labels  = (const int*)  d_in[1];
    const float* centers = (const float*)d_in[2];
    float*       out     = (float*)d_out;
    float*       bsums   = (float*)d_ws;          // 256 floats of scratch

    const int nBlocks = BATCH / 16;               // 256

    CenterLoss_dist_kernel<<<nBlocks, 256, 0, stream>>>(x, labels, centers, bsums);
    CenterLoss_reduce_kernel<<<1, 256, 0, stream>>>(bsums, out);
}